// SoftMinChamfer_67413806678293
// MI455X (gfx1250) — compile-verified
//
#include <hip/hip_runtime.h>
#include <math.h>

typedef __attribute__((ext_vector_type(2))) float v2f;
typedef __attribute__((ext_vector_type(8))) float v8f;

#define NPTS   4096
#define TILE   16
#define NTILES (NPTS / TILE)   // 256 column tiles
#define WAVES  8               // 256 threads = 8 wave32

// Work in base-2 logits: l2 = -(1/T)*log2(e) * D, so v_exp_f32 (which is
// exp2) is used directly with no pre-multiply. Softmax is base-invariant.
#define NEG_L2E_OVER_T (-144.26950408889634f)   // -100 * log2(e)

// Robust merge of two online-softmax stat triples (cold path only).
__device__ inline void merge_stat(float& m, float& s, float& t,
                                  float mo, float so, float to) {
    float nm = fmaxf(m, mo);
    float e1 = __builtin_amdgcn_exp2f(m - nm);   // exp2(-inf) -> 0
    float e2 = __builtin_amdgcn_exp2f(mo - nm);
    s = s * e1 + so * e2;
    t = t * e1 + to * e2;
    m = nm;
}

// One workgroup = one (problem, 16-row tile).
// problem p in [0,2B): b = p>>1, p&1 selects x/y swap (prec vs cov leg).
// For each row i of the tile we compute sum_j softmax_j(-D/T)*D via
// online softmax over all 4096 columns. The K=3 dot products come from
// v_wmma_f32_16x16x4_f32 (dim padded to 4), with A pre-scaled by -2 and
// C preloaded with |x|^2 + |y|^2 so the matrix op emits D^2 directly.
__global__ __launch_bounds__(256) void softmin_rows_kernel(
    const float* __restrict__ x, const float* __restrict__ y,
    float* __restrict__ partials)
{
    const int p    = blockIdx.y;
    const int b    = p >> 1;
    const int swap = p & 1;
    const float* abase = (swap ? y : x) + (size_t)b * NPTS * 3;  // row points
    const float* bbase = (swap ? x : y) + (size_t)b * NPTS * 3;  // col points

    const int lane = threadIdx.x & 31;
    const int wave = threadIdx.x >> 5;
    const int half = (lane >> 4) & 1;   // 0: K=0,1 lanes; 1: K=2,pad lanes
    const int l16  = lane & 15;

    // ---- A operand: 16 rows of this tile, ISA 16x4 f32 layout, scaled -2 ----
    const int r  = blockIdx.x * TILE + l16;
    const float ax = abase[r * 3 + 0];
    const float ay = abase[r * 3 + 1];
    const float az = abase[r * 3 + 2];
    const float anorm_mine = ax * ax + ay * ay + az * az;
    v2f aop;
    aop.x = -2.0f * (half ? az : ax);
    aop.y = half ? 0.0f : (-2.0f * ay);

    // Each lane's accumulator covers rows 8*half + v; fetch those norms.
    float xn[8];
#pragma unroll
    for (int v = 0; v < 8; ++v) {
        int src = half ? (24 + v) : v;           // a lane holding row 8*half+v
        xn[v] = __shfl(anorm_mine, src, 32);
    }

    float m[8], s[8], t[8];
#pragma unroll
    for (int v = 0; v < 8; ++v) {
        m[v] = -__builtin_inff(); s[v] = 0.0f; t[v] = 0.0f;
    }

    // ---- column-tile loop: each wave takes every 8th tile (uniform EXEC).
    // Unroll x2 so two tiles' TRANS (sqrt/exp), VALU and WMMA streams
    // interleave and fill the trans co-execution hazard slots.
#pragma unroll 2
    for (int jt = wave; jt < NTILES; jt += WAVES) {
        const int c  = jt * TILE + l16;
        const float bx = bbase[c * 3 + 0];
        const float by = bbase[c * 3 + 1];
        const float bz = bbase[c * 3 + 2];
        const float bn = bx * bx + by * by + bz * bz;
        v2f bop;
        bop.x = half ? bz : bx;
        bop.y = half ? 0.0f : by;

        // C = |x_row|^2 + |y_col|^2 so D = A(-2x) * B(y) + C = squared dist
        v8f cin;
#pragma unroll
        for (int v = 0; v < 8; ++v) cin[v] = xn[v] + bn;

        v8f acc = __builtin_amdgcn_wmma_f32_16x16x4_f32(
            /*neg_a=*/false, aop, /*neg_b=*/false, bop,
            /*c_mod=*/(short)0, cin, /*reuse_a=*/false, /*reuse_b=*/false);

#pragma unroll
        for (int v = 0; v < 8; ++v) {
            // |.| instead of max(0,.): negative values are only cancellation
            // noise ~0, and fabs folds into the sqrt source modifier.
            float D  = __builtin_amdgcn_sqrtf(__builtin_fabsf(acc[v]));
            float l  = NEG_L2E_OVER_T * D;    // base-2 logit
            float d  = m[v] - l;              // >= 0: max unchanged; < 0: new max
            float e  = __builtin_amdgcn_exp2f(-__builtin_fabsf(d));
            bool  grow = d < 0.0f;
            s[v] = grow ? (s[v] * e + 1.0f) : (s[v] + e);
            t[v] = grow ? (t[v] * e + D)    : (t[v] + D * e);
            m[v] = grow ? l : m[v];           // reuse compare, no canonicalize
        }
    }

    // ---- reduce across the 16 columns held by each 16-lane group ----
#pragma unroll
    for (int mask = 1; mask < 16; mask <<= 1) {
#pragma unroll
        for (int v = 0; v < 8; ++v) {
            float mo = __shfl_xor(m[v], mask, 32);
            float so = __shfl_xor(s[v], mask, 32);
            float to = __shfl_xor(t[v], mask, 32);
            merge_stat(m[v], s[v], t[v], mo, so, to);
        }
    }

    // ---- merge across waves via LDS ----
    __shared__ float lds[WAVES][TILE][3];
    if (l16 == 0) {
#pragma unroll
        for (int v = 0; v < 8; ++v) {
            int row = half * 8 + v;
            lds[wave][row][0] = m[v];
            lds[wave][row][1] = s[v];
            lds[wave][row][2] = t[v];
        }
    }
    __syncthreads();

    __shared__ float rowvals[TILE];
    if (threadIdx.x < TILE) {
        int row = threadIdx.x;
        float M = -__builtin_inff(), S = 0.0f, T = 0.0f;
        for (int w = 0; w < WAVES; ++w)
            merge_stat(M, S, T, lds[w][row][0], lds[w][row][1], lds[w][row][2]);
        rowvals[row] = T / S;   // = sum_j softmax_j * D for this row
    }
    __syncthreads();

    if (threadIdx.x == 0) {
        float sum = 0.0f;
#pragma unroll
        for (int row = 0; row < TILE; ++row) sum += rowvals[row];
        partials[blockIdx.y * gridDim.x + blockIdx.x] = sum;
    }
}

// Deterministic fixed-order final reduction of the per-workgroup partials.
__global__ __launch_bounds__(256) void final_reduce_kernel(
    const float* __restrict__ partials, float* __restrict__ out,
    int n, float scale)
{
    __shared__ float sm[256];
    float s = 0.0f;
    for (int i = threadIdx.x; i < n; i += 256) s += partials[i];
    sm[threadIdx.x] = s;
    __syncthreads();
    for (int off = 128; off > 0; off >>= 1) {
        if (threadIdx.x < off) sm[threadIdx.x] += sm[threadIdx.x + off];
        __syncthreads();
    }
    if (threadIdx.x == 0) out[0] = sm[0] * scale;
}

extern "C" void kernel_launch(void* const* d_in, const int* in_sizes, int n_in,
                              void* d_out, int out_size, void* d_ws, size_t ws_size,
                              hipStream_t stream) {
    const float* x = (const float*)d_in[0];
    const float* y = (const float*)d_in[1];
    float* partials = (float*)d_ws;

    const int batch = in_sizes[0] / (NPTS * 3);   // 4 for the reference shapes
    const int nprob = batch * 2;                  // prec + cov per batch

    dim3 grid(NTILES, nprob);
    softmin_rows_kernel<<<grid, 256, 0, stream>>>(x, y, partials);

    // out = (sum over all 2B problems of row sums) / (B * NPTS)
    const float scale = 1.0f / ((float)batch * (float)NPTS);
    final_reduce_kernel<<<1, 256, 0, stream>>>(partials, (float*)d_out,
                                               NTILES * nprob, scale);
}